// mwGCN_20366734917713
// MI455X (gfx1250) — compile-verified
//
#include <hip/hip_runtime.h>

#define R_    3
#define N_    50000
#define E_    800000
#define DIN   128
#define DOUT  64
#define TILES 5          // row tiles (16 rows each) per block; 50000 / (16*5) = 625

typedef __attribute__((ext_vector_type(2))) float v2f;
typedef __attribute__((ext_vector_type(8))) float v8f;

// ---------------------------------------------------------------------------
// Kernel 1: support[r] = X[r] @ W[r] via V_WMMA_F32_16X16X4_F32 (fp32 WMMA).
// One wave = one 16-column slice (n0 = wave*16). B fragments for the whole
// K=128 dimension are preloaded into 64 VGPRs and reused across TILES row
// tiles, so the steady state is 1 global_load_b64 (A) per WMMA.
// ---------------------------------------------------------------------------
__global__ __launch_bounds__(128) void gcn_gemm_wmma(const float* __restrict__ X,
                                                     const float* __restrict__ W,
                                                     float* __restrict__ support) {
    const int r    = blockIdx.y;
    const int row0 = blockIdx.x * (16 * TILES);
    const int wave = threadIdx.x >> 5;
    const int lane = threadIdx.x & 31;
    const int n0   = wave << 4;                // column tile within DOUT
    const int mn   = lane & 15;                // M for A-frag, N for B/C-frag
    const int koff = (lane >> 4) << 1;         // lanes 0-15 -> K{0,1}, 16-31 -> K{2,3}

    // ---- preload all B fragments (K = 0..127) into registers -------------
    const float* __restrict__ Wb = W + (size_t)r * DIN * DOUT + n0 + mn;
    v2f b[32];
    #pragma unroll
    for (int k = 0; k < 32; ++k) {
        b[k].x = Wb[(size_t)(4 * k + koff)     * DOUT];
        b[k].y = Wb[(size_t)(4 * k + koff + 1) * DOUT];
    }

    const float* __restrict__ Xa =
        X + ((size_t)r * N_ + row0) * DIN + (size_t)mn * DIN + koff;
    float* __restrict__ S =
        support + ((size_t)r * N_ + row0) * DOUT + n0 + mn;
    const int mbase = (lane >> 4) << 3;        // C/D: lanes 16-31 hold M = v+8

    for (int t = 0; t < TILES; ++t) {
        v8f acc = {0.f, 0.f, 0.f, 0.f, 0.f, 0.f, 0.f, 0.f};
        const float* __restrict__ At = Xa + (size_t)t * 16 * DIN;
        #pragma unroll
        for (int k = 0; k < 32; ++k) {
            v2f a = *(const v2f*)(At + 4 * k);   // A[mn][4k+koff .. +1], 8B load
            acc = __builtin_amdgcn_wmma_f32_16x16x4_f32(
                false, a, false, b[k], (short)0, acc, false, false);
        }
        #pragma unroll
        for (int v = 0; v < 8; ++v)
            S[(size_t)(t * 16 + mbase + v) * DOUT] = acc[v];
    }
}

// ---------------------------------------------------------------------------
// Kernel 2a: out[r][n][o] = biases[r][o]
// ---------------------------------------------------------------------------
__global__ void gcn_init_out(const float* __restrict__ biases,
                             float* __restrict__ out) {
    size_t idx = (size_t)blockIdx.x * blockDim.x + threadIdx.x;
    const size_t total = (size_t)R_ * N_ * DOUT;
    if (idx >= total) return;
    int o = (int)(idx & (DOUT - 1));
    int r = (int)(idx / ((size_t)N_ * DOUT));
    out[idx] = biases[r * DOUT + o];
}

// ---------------------------------------------------------------------------
// Kernel 2b: COO SpMM scatter. 16 threads/edge, float4 per thread.
// Gather from L2-resident support, global_atomic_add_f32 scatter into out.
// ---------------------------------------------------------------------------
__global__ __launch_bounds__(256) void gcn_spmm_scatter(const int* __restrict__ rows,
                                                        const int* __restrict__ cols,
                                                        const float* __restrict__ vals,
                                                        const float* __restrict__ support,
                                                        float* __restrict__ out) {
    const int r = blockIdx.y;
    const int e = blockIdx.x * 16 + (threadIdx.x >> 4);   // E_ % 16 == 0
    if (e >= E_) return;
    const int sub = (threadIdx.x & 15) << 2;              // 4 floats per thread

    const size_t eb  = (size_t)r * E_ + e;
    const int   row  = rows[eb];
    const int   col  = cols[eb];
    const float val  = vals[eb];

    const float4 sv = *(const float4*)(support + ((size_t)r * N_ + col) * DOUT + sub);
    float* dst = out + ((size_t)r * N_ + row) * DOUT + sub;
    atomicAdd(dst + 0, val * sv.x);
    atomicAdd(dst + 1, val * sv.y);
    atomicAdd(dst + 2, val * sv.z);
    atomicAdd(dst + 3, val * sv.w);
}

// ---------------------------------------------------------------------------
// Kernel 3: scores -> softmax over R=3 -> weighted combine; emits both
// real_output (N x 64) and w.T (N x 3) into d_out (concatenated).
// One 64-thread block per node (thread o = output column o).
// ---------------------------------------------------------------------------
__global__ __launch_bounds__(64) void gcn_attention(const float* __restrict__ out,
                                                    const float* __restrict__ att_w,
                                                    const float* __restrict__ att_b,
                                                    float* __restrict__ dout) {
    const int n = blockIdx.x;
    const int o = threadIdx.x;

    __shared__ float red[R_][DOUT];
    __shared__ float sc[R_];
    __shared__ float wsm[R_];

    float ov[R_];
    const float aw = att_w[o];
    #pragma unroll
    for (int r = 0; r < R_; ++r) {
        ov[r] = out[((size_t)r * N_ + n) * DOUT + o];
        red[r][o] = ov[r] * aw;
    }
    __syncthreads();

    if (o < R_) {
        float s = att_b[0];
        #pragma unroll 8
        for (int i = 0; i < DOUT; ++i) s += red[o][i];
        sc[o] = s;
    }
    __syncthreads();

    if (o == 0) {
        float m  = fmaxf(sc[0], fmaxf(sc[1], sc[2]));
        float e0 = __expf(sc[0] - m), e1 = __expf(sc[1] - m), e2 = __expf(sc[2] - m);
        float inv = 1.f / (e0 + e1 + e2);
        wsm[0] = e0 * inv; wsm[1] = e1 * inv; wsm[2] = e2 * inv;
        float* wT = dout + (size_t)N_ * DOUT + (size_t)n * R_;
        wT[0] = wsm[0]; wT[1] = wsm[1]; wT[2] = wsm[2];
    }
    __syncthreads();

    dout[(size_t)n * DOUT + o] = wsm[0] * ov[0] + wsm[1] * ov[1] + wsm[2] * ov[2];
}

// ---------------------------------------------------------------------------
extern "C" void kernel_launch(void* const* d_in, const int* in_sizes, int n_in,
                              void* d_out, int out_size, void* d_ws, size_t ws_size,
                              hipStream_t stream) {
    const int*   adj_rows = (const int*)d_in[0];
    const int*   adj_cols = (const int*)d_in[1];
    const float* adj_vals = (const float*)d_in[2];
    const float* X        = (const float*)d_in[3];
    const float* W        = (const float*)d_in[4];
    const float* biases   = (const float*)d_in[5];
    const float* att_w    = (const float*)d_in[6];
    const float* att_b    = (const float*)d_in[7];
    float* dout = (float*)d_out;

    float* support = (float*)d_ws;                        // R*N*DOUT floats (38.4 MB)
    float* outbuf  = support + (size_t)R_ * N_ * DOUT;    // R*N*DOUT floats (38.4 MB)

    // 1) dense transform via fp32 WMMA (B fragments register-resident)
    dim3 g1(N_ / (16 * TILES), R_);
    gcn_gemm_wmma<<<g1, 128, 0, stream>>>(X, W, support);

    // 2) out = bias, then atomic COO scatter
    const size_t total = (size_t)R_ * N_ * DOUT;
    gcn_init_out<<<(unsigned)((total + 255) / 256), 256, 0, stream>>>(biases, outbuf);
    dim3 g2(E_ / 16, R_);
    gcn_spmm_scatter<<<g2, 256, 0, stream>>>(adj_rows, adj_cols, adj_vals, support, outbuf);

    // 3) attention combine -> d_out (real_output then w.T)
    gcn_attention<<<N_, 64, 0, stream>>>(outbuf, att_w, att_b, dout);
}